// GraphEncoder_35948876268152
// MI455X (gfx1250) — compile-verified
//
#include <hip/hip_runtime.h>
#include <hip/hip_bf16.h>

// ---------------------------------------------------------------------------
// 2-layer GCN + mean-pool + FC for MI455X (gfx1250, wave32, WMMA).
// GEMMs use V_WMMA_F32_16X16X4_F32 (exact f32 path) with the weight matrix
// staged in LDS once per block (8 waves share it, bank-conflict-free reads).
// Aggregation is the bandwidth-bound part: ~3-4 GB of gather + f32 atomics.
// ---------------------------------------------------------------------------

#define N_NODES_C 100000
#define N_EDGES_C 3200000
#define IN_DIM_C  128
#define HID_C     64
#define OUT_DIM_C 128
#define N_GRAPHS_C 8

typedef __attribute__((ext_vector_type(2))) float v2f;
typedef __attribute__((ext_vector_type(8))) float v8f;

static inline int cdiv(int a, int b) { return (a + b - 1) / b; }

// ------------------------- degree / normalization --------------------------

__global__ void deg_init_kernel(float* deg, int n) {
    int t = blockIdx.x * blockDim.x + threadIdx.x;
    if (t < n) deg[t] = 1.0f;   // self-loop weight
}

__global__ void deg_accum_kernel(const int* __restrict__ dst,
                                 const float* __restrict__ w,
                                 float* deg, int E) {
    int e = blockIdx.x * blockDim.x + threadIdx.x;
    if (e < E) atomicAdd(&deg[dst[e]], w[e]);
}

__global__ void make_dinv_kernel(float* deg, int n) {
    int t = blockIdx.x * blockDim.x + threadIdx.x;
    if (t < n) {
        float d = deg[t];
        deg[t] = (d > 0.0f) ? rsqrtf(d) : 0.0f;   // in-place dinv
    }
}

__global__ void edge_norm_kernel(const int* __restrict__ src,
                                 const int* __restrict__ dst,
                                 const float* __restrict__ w,
                                 const float* __restrict__ dinv,
                                 float* __restrict__ norm, int E) {
    int e = blockIdx.x * blockDim.x + threadIdx.x;
    if (e < E) norm[e] = dinv[src[e]] * w[e] * dinv[dst[e]];
}

// ------------------------------ WMMA GEMM ----------------------------------
// C[M x 64] = A[M x K] @ B[K x 64].  M % 16 == 0, K % 4 == 0, K <= 128.
// Block = 256 threads = 8 waves; each wave computes one 16x64 output slab
// with 4 v8f accumulators, stepping K by 4 through V_WMMA_F32_16X16X4_F32.
// B is staged in LDS once per block. LDS row stride = 80 floats so the two
// half-wave K-rows (ka vs ka+2) hit disjoint bank windows: bank = (16*ka+col)%64.
// A frag (16x4 f32, 2 VGPRs): lanes 0-15 -> K = {k,k+1}; lanes 16-31 -> {k+2,k+3}.
// B frag (4x16 f32, 2 VGPRs): same K split, N = lane & 15.
// C/D (16x16 f32, 8 VGPRs): vgpr i holds row i (lanes 0-15) / row i+8 (lanes 16-31).

#define BSTRIDE 80
#define GEMM_TB 256
#define WAVES_PER_BLOCK (GEMM_TB / 32)

__global__ __launch_bounds__(GEMM_TB)
void gemm_wmma_n64_kernel(const float* __restrict__ A,
                          const float* __restrict__ B,
                          float* __restrict__ C,
                          int M, int K) {
    __shared__ float sB[IN_DIM_C * BSTRIDE];     // 128*80*4 = 40 KB (max K)

    // cooperative fill of B into LDS (float4 granularity), before any wave exits
    int tid = threadIdx.x;
    for (int i = tid; i < K * 16; i += GEMM_TB) {
        int krow = i >> 4;                        // 0..K-1
        int c4   = (i & 15) << 2;                 // 0..60 step 4
        float4 v = *reinterpret_cast<const float4*>(B + (size_t)krow * 64 + c4);
        *reinterpret_cast<float4*>(&sB[krow * BSTRIDE + c4]) = v;
    }
    __syncthreads();

    int wave  = blockIdx.x * WAVES_PER_BLOCK + (tid >> 5);
    int lane  = tid & 31;
    int ntile = M >> 4;
    if (wave >= ntile) return;                    // uniform per-wave: EXEC all-1s

    int row0 = wave << 4;
    int r    = lane & 15;                         // row/col-within-tile
    int h    = lane >> 4;                         // 0 or 1 (K-half select)

    v8f acc0 = {}, acc1 = {}, acc2 = {}, acc3 = {};
    const float* arow = A + (size_t)(row0 + r) * K;

    for (int k = 0; k < K; k += 4) {
        int ka = k + 2 * h;
        v2f a;
        a.x = arow[ka];
        a.y = arow[ka + 1];

        const float* s0 = &sB[ka * BSTRIDE];
        const float* s1 = s0 + BSTRIDE;
        v2f bb0, bb1, bb2, bb3;
        bb0.x = s0[r];      bb0.y = s1[r];
        bb1.x = s0[16 + r]; bb1.y = s1[16 + r];
        bb2.x = s0[32 + r]; bb2.y = s1[32 + r];
        bb3.x = s0[48 + r]; bb3.y = s1[48 + r];

        acc0 = __builtin_amdgcn_wmma_f32_16x16x4_f32(false, a, false, bb0, (short)0, acc0, false, false);
        acc1 = __builtin_amdgcn_wmma_f32_16x16x4_f32(false, a, false, bb1, (short)0, acc1, false, false);
        acc2 = __builtin_amdgcn_wmma_f32_16x16x4_f32(false, a, false, bb2, (short)0, acc2, false, false);
        acc3 = __builtin_amdgcn_wmma_f32_16x16x4_f32(false, a, false, bb3, (short)0, acc3, false, false);
    }

    float* crow = C + (size_t)(row0 + 8 * h) * 64 + r;
    #pragma unroll
    for (int i = 0; i < 8; ++i) {
        float* cp = crow + (size_t)i * 64;
        cp[0]  = acc0[i];
        cp[16] = acc1[i];
        cp[32] = acc2[i];
        cp[48] = acc3[i];
    }
}

// ------------------------------ aggregation --------------------------------

// out[n][c] = h[n][c] * dinv[n]^2   (self-loop contribution, initializes acc)
__global__ void agg_init_kernel(const float* __restrict__ h,
                                const float* __restrict__ dinv,
                                float* __restrict__ out, int n) {
    int t = blockIdx.x * blockDim.x + threadIdx.x;   // over n*64
    if (t >= n * HID_C) return;
    int node = t >> 6;
    float s = dinv[node];
    out[t] = h[t] * (s * s);
}

// 16 threads per edge, 4 channels each (float4 gather + 4 f32 atomics).
__global__ void edge_scatter_kernel(const int* __restrict__ src,
                                    const int* __restrict__ dst,
                                    const float* __restrict__ norm,
                                    const float* __restrict__ h,
                                    float* __restrict__ out, int E) {
    int t = blockIdx.x * blockDim.x + threadIdx.x;
    int e = t >> 4;
    if (e >= E) return;
    int c4 = (t & 15) << 2;
    float nm = norm[e];
    int s = src[e], d = dst[e];
    const float4 v = *reinterpret_cast<const float4*>(h + (size_t)s * HID_C + c4);
    float* op = out + (size_t)d * HID_C + c4;
    atomicAdd(op + 0, v.x * nm);
    atomicAdd(op + 1, v.y * nm);
    atomicAdd(op + 2, v.z * nm);
    atomicAdd(op + 3, v.w * nm);
}

__global__ void bias_relu_kernel(const float* __restrict__ in,
                                 const float* __restrict__ b,
                                 float* __restrict__ out, int n) {
    int t = blockIdx.x * blockDim.x + threadIdx.x;   // over n*64
    if (t >= n * HID_C) return;
    float v = in[t] + b[t & 63];
    out[t] = v > 0.0f ? v : 0.0f;
}

// ------------------------------- pooling -----------------------------------

__global__ void pool_zero_kernel(float* sums, float* counts) {
    int t = blockIdx.x * blockDim.x + threadIdx.x;
    if (t < N_GRAPHS_C * HID_C) sums[t] = 0.0f;
    if (t < N_GRAPHS_C) counts[t] = 0.0f;
}

// per-node: add (agg + b2) into sums[batch[node]]; count nodes per graph.
__global__ void pool_accum_kernel(const float* __restrict__ h,
                                  const float* __restrict__ b2,
                                  const int* __restrict__ batch,
                                  float* sums, float* counts, int n) {
    int t = blockIdx.x * blockDim.x + threadIdx.x;   // over n*16
    int node = t >> 4;
    if (node >= n) return;
    int c4 = (t & 15) << 2;
    int g = batch[node];
    const float4 v = *reinterpret_cast<const float4*>(h + (size_t)node * HID_C + c4);
    float* sp = sums + g * HID_C + c4;
    atomicAdd(sp + 0, v.x + b2[c4 + 0]);
    atomicAdd(sp + 1, v.y + b2[c4 + 1]);
    atomicAdd(sp + 2, v.z + b2[c4 + 2]);
    atomicAdd(sp + 3, v.w + b2[c4 + 3]);
    if ((t & 15) == 0) atomicAdd(&counts[g], 1.0f);
}

// out[8 x 128] = (sums/counts) @ Wfc + bfc  -- tiny, single block.
__global__ void final_fc_kernel(const float* __restrict__ sums,
                                const float* __restrict__ counts,
                                const float* __restrict__ Wfc,
                                const float* __restrict__ bfc,
                                float* __restrict__ out) {
    int j = threadIdx.x;                              // 0..127
    if (j >= OUT_DIM_C) return;
    for (int g = 0; g < N_GRAPHS_C; ++g) {
        float c = counts[g];
        float inv = 1.0f / (c > 1.0f ? c : 1.0f);
        float acc = 0.0f;
        for (int k = 0; k < HID_C; ++k)
            acc += (sums[g * HID_C + k] * inv) * Wfc[k * OUT_DIM_C + j];
        out[g * OUT_DIM_C + j] = acc + bfc[j];
    }
}

// ------------------------------- launcher ----------------------------------

extern "C" void kernel_launch(void* const* d_in, const int* in_sizes, int n_in,
                              void* d_out, int out_size, void* d_ws, size_t ws_size,
                              hipStream_t stream) {
    const float* x      = (const float*)d_in[0];   // [n,128]
    const int*   eidx   = (const int*)  d_in[1];   // [2,E]
    const float* ew     = (const float*)d_in[2];   // [E]
    const int*   batch  = (const int*)  d_in[3];   // [n]
    const float* W1     = (const float*)d_in[4];   // [128,64]
    const float* b1     = (const float*)d_in[5];   // [64]
    const float* W2     = (const float*)d_in[6];   // [64,64]
    const float* b2     = (const float*)d_in[7];   // [64]
    const float* Wfc    = (const float*)d_in[8];   // [64,128]
    const float* bfc    = (const float*)d_in[9];   // [128]
    float* out = (float*)d_out;

    const int n = in_sizes[0] / IN_DIM_C;          // 100000
    const int E = in_sizes[1] / 2;                 // 3200000
    const int* src = eidx;
    const int* dst = eidx + E;

    // workspace layout (floats)
    float* ws     = (float*)d_ws;
    float* dinv   = ws;                            // [n]      (deg then dinv)
    float* bufA   = dinv + n;                      // [n*64]
    float* bufB   = bufA + (size_t)n * HID_C;      // [n*64]
    float* norm   = bufB + (size_t)n * HID_C;      // [E]
    float* sums   = norm + E;                      // [8*64]
    float* counts = sums + N_GRAPHS_C * HID_C;     // [8]

    const int TB = 256;

    // normalization coefficients
    deg_init_kernel  <<<cdiv(n, TB), TB, 0, stream>>>(dinv, n);
    deg_accum_kernel <<<cdiv(E, TB), TB, 0, stream>>>(dst, ew, dinv, E);
    make_dinv_kernel <<<cdiv(n, TB), TB, 0, stream>>>(dinv, n);
    edge_norm_kernel <<<cdiv(E, TB), TB, 0, stream>>>(src, dst, ew, dinv, norm, E);

    const int ntile = n >> 4;                      // 6250 row tiles of 16

    // layer 1: h1 = x @ W1  (WMMA), then normalized aggregation
    gemm_wmma_n64_kernel<<<cdiv(ntile, WAVES_PER_BLOCK), GEMM_TB, 0, stream>>>(x, W1, bufA, n, IN_DIM_C);
    agg_init_kernel     <<<cdiv(n * HID_C, TB), TB, 0, stream>>>(bufA, dinv, bufB, n);
    edge_scatter_kernel <<<cdiv(E * 16, TB), TB, 0, stream>>>(src, dst, norm, bufA, bufB, E);
    bias_relu_kernel    <<<cdiv(n * HID_C, TB), TB, 0, stream>>>(bufB, b1, bufA, n);

    // layer 2: h3 = relu(out1) @ W2  (WMMA), then normalized aggregation
    gemm_wmma_n64_kernel<<<cdiv(ntile, WAVES_PER_BLOCK), GEMM_TB, 0, stream>>>(bufA, W2, bufB, n, HID_C);
    agg_init_kernel     <<<cdiv(n * HID_C, TB), TB, 0, stream>>>(bufB, dinv, bufA, n);
    edge_scatter_kernel <<<cdiv(E * 16, TB), TB, 0, stream>>>(src, dst, norm, bufB, bufA, E);

    // mean-pool over batch ids + final FC
    pool_zero_kernel  <<<cdiv(N_GRAPHS_C * HID_C, TB), TB, 0, stream>>>(sums, counts);
    pool_accum_kernel <<<cdiv(n * 16, TB), TB, 0, stream>>>(bufA, b2, batch, sums, counts, n);
    final_fc_kernel   <<<1, OUT_DIM_C, 0, stream>>>(sums, counts, Wfc, bfc, out);
}